// InteractionBlock_13615046328447
// MI455X (gfx1250) — compile-verified
//
#include <hip/hip_runtime.h>

typedef float v2f __attribute__((ext_vector_type(2)));
typedef float v8f __attribute__((ext_vector_type(8)));

#define PAD 68   // row pitch (floats) for 16x64 LDS tiles, padded vs 64 banks

__device__ __forceinline__ v8f wmma_f32(v2f a, v2f b, v8f c) {
  // V_WMMA_F32_16X16X4_F32 : D = A(16x4) * B(4x16) + C(16x16)
  return __builtin_amdgcn_wmma_f32_16x16x4_f32(false, a, false, b, (short)0, c,
                                               false, false);
}

__global__ void zero_kernel(float* __restrict__ p, size_t n) {
  size_t i = (size_t)blockIdx.x * blockDim.x + threadIdx.x;
  if (i < n) p[i] = 0.0f;
}

// Repack W (K x 64, row-major) into B-fragment order with a folded scale:
// WP[p][col] = {scale*W[2p][col], scale*W[2p+1][col]}  (p = 0..K/2-1)
// A WMMA B-fragment then is ONE b64 load: WP[(kk*2 + lanehalf)*64 + col].
__global__ void perm_kernel(const float* __restrict__ W, float2* __restrict__ WP,
                            int halfK, float scale) {
  int i = blockIdx.x * blockDim.x + threadIdx.x;
  if (i < halfK * 64) {
    int p = i >> 6, col = i & 63;
    WP[i] = make_float2(scale * W[(2 * p) * 64 + col],
                        scale * W[(2 * p + 1) * 64 + col]);
  }
}

// ---------------------------------------------------------------------------
// Edge kernel: one block (128 thr = 4 waves) per 16-edge tile.
// Scales folded into permuted weights: Wap=A_TP*Wa, Wbp=A_TP/sqrt3*Wb,
// Wcp=A_TP*Wc, Wdp=A_TP*Wd.
// ---------------------------------------------------------------------------
__global__ __launch_bounds__(128)
void edge_kernel(const float* __restrict__ node_feats,
                 const float* __restrict__ edge_feats,
                 const long long* __restrict__ edge_index,
                 const float2* __restrict__ Wap, const float2* __restrict__ Wbp,
                 const float2* __restrict__ Wcp, const float2* __restrict__ Wdp,
                 float* __restrict__ aggS, float* __restrict__ aggV,
                 int N, int E) {
  __shared__ float S1[16 * PAD], VX[16 * PAD], VY[16 * PAD], VZ[16 * PAD];
  __shared__ float AA[16 * PAD], BB[16 * PAD];
  __shared__ float sEs[16], sEv[16][3];
  __shared__ int   sSend[16], sRecv[16];

  const int tid = threadIdx.x;
  const int e0  = blockIdx.x * 16;

  if (tid < 16) {
    int e = e0 + tid;
    if (e < E) {
      sSend[tid]  = (int)edge_index[e];
      sRecv[tid]  = (int)edge_index[(size_t)E + e];
      sEs[tid]    = edge_feats[e * 4 + 0];
      sEv[tid][0] = edge_feats[e * 4 + 1];
      sEv[tid][1] = edge_feats[e * 4 + 2];
      sEv[tid][2] = edge_feats[e * 4 + 3];
    } else {
      sSend[tid] = 0; sRecv[tid] = -1;
      sEs[tid] = 0.0f; sEv[tid][0] = sEv[tid][1] = sEv[tid][2] = 0.0f;
    }
  }
  __syncthreads();

  // Stage gathered sender features + per-edge-scaled GEMM inputs into LDS.
  #pragma unroll
  for (int it = 0; it < 8; ++it) {
    int idx = it * 128 + tid;          // 16 rows x 64 channels
    int row = idx >> 6, ch = idx & 63;
    const float* nf = node_feats + (size_t)sSend[row] * 256;
    float s  = nf[ch];
    float vx = nf[64 + ch * 3 + 0];
    float vy = nf[64 + ch * 3 + 1];
    float vz = nf[64 + ch * 3 + 2];
    float es = sEs[row];
    float b  = vx * sEv[row][0] + vy * sEv[row][1] + vz * sEv[row][2];
    S1[row * PAD + ch] = s;
    VX[row * PAD + ch] = vx;
    VY[row * PAD + ch] = vy;
    VZ[row * PAD + ch] = vz;
    AA[row * PAD + ch] = es * s;   // -> Wap (A_TP folded in weights)
    BB[row * PAD + ch] = b;        // -> Wbp (A_TP/sqrt3 folded in weights)
  }
  __syncthreads();

  const int wave  = tid >> 5;                 // column tile 0..3
  const int lane  = tid & 31;
  const int mrow  = lane & 15;                // A/C row (low half)
  const int khalf = (lane >> 4) * 2;          // 0 or 2 : K offset within step
  const int col   = wave * 16 + (lane & 15);  // output column
  const int bbase = ((lane >> 4) << 6) + col; // B frag base (float2 elements)

  v8f acc_s = {0}, acc_q = {0}, acc_x = {0}, acc_y = {0}, acc_z = {0};

  #pragma unroll 4
  for (int kk = 0; kk < 16; ++kk) {
    const int ab = mrow * PAD + kk * 4 + khalf;
    v2f fa = {AA[ab], AA[ab + 1]};
    v2f fb = {BB[ab], BB[ab + 1]};
    v2f fs = {S1[ab], S1[ab + 1]};
    v2f fx = {VX[ab], VX[ab + 1]};
    v2f fy = {VY[ab], VY[ab + 1]};
    v2f fz = {VZ[ab], VZ[ab + 1]};
    float2 wa = Wap[bbase + kk * 128];
    float2 wb = Wbp[bbase + kk * 128];
    float2 wc = Wcp[bbase + kk * 128];
    float2 wd = Wdp[bbase + kk * 128];
    v2f ba = {wa.x, wa.y};
    v2f bb = {wb.x, wb.y};
    v2f bc = {wc.x, wc.y};
    v2f bd = {wd.x, wd.y};
    acc_s = wmma_f32(fa, ba, acc_s);   // mid_s  (Wa + Wb fused in C)
    acc_s = wmma_f32(fb, bb, acc_s);
    acc_q = wmma_f32(fs, bc, acc_q);   // s1 @ Wc
    acc_x = wmma_f32(fx, bd, acc_x);   // v_i @ Wd
    acc_y = wmma_f32(fy, bd, acc_y);
    acc_z = wmma_f32(fz, bd, acc_z);
  }

  // Scatter: segment-sum via fp32 atomics (L2-resident agg planes).
  const size_t NW = (size_t)N * 64;
  #pragma unroll
  for (int r = 0; r < 8; ++r) {
    int rowM = r + ((lane >> 4) << 3);
    int recv = sRecv[rowM];
    if (recv >= 0) {
      float es  = sEs[rowM];
      float mvx = sEv[rowM][0] * acc_q[r] + es * acc_x[r];
      float mvy = sEv[rowM][1] * acc_q[r] + es * acc_y[r];
      float mvz = sEv[rowM][2] * acc_q[r] + es * acc_z[r];
      size_t base = (size_t)recv * 64 + col;
      atomicAdd(&aggS[base],          acc_s[r]);
      atomicAdd(&aggV[base],          mvx);
      atomicAdd(&aggV[NW + base],     mvy);
      atomicAdd(&aggV[2 * NW + base], mvz);
    }
  }
}

// ---------------------------------------------------------------------------
// Node kernel: one block (128 thr) per 16-node tile.
// Agg planes staged global->LDS with CDNA5 async-to-LDS (ASYNCcnt) copies.
// GEMM1: x = agg @ (A_LIN*Wls/Wlv)        (K=64)
// GEMM2: skip = (x (x) attrs) @ (A_SKIP*Wss/Wsv)   (K=256, A built on the fly)
// ---------------------------------------------------------------------------
__global__ __launch_bounds__(128)
void node_kernel(const float* __restrict__ node_attrs,
                 const float2* __restrict__ Wlsp, const float2* __restrict__ Wlvp,
                 const float2* __restrict__ Wssp, const float2* __restrict__ Wsvp,
                 const float* __restrict__ aggS, const float* __restrict__ aggV,
                 float* __restrict__ out, int N) {
  __shared__ float AS[16 * PAD], AV0[16 * PAD], AV1[16 * PAD], AV2[16 * PAD];
  __shared__ float XS[16 * PAD], XV0[16 * PAD], XV1[16 * PAD], XV2[16 * PAD];
  __shared__ float AT[16][4];

  const int tid = threadIdx.x;
  const int n0  = blockIdx.x * 16;
  const size_t NW = (size_t)N * 64;

  if (tid < 64) {
    int row = tid >> 2, v = tid & 3;
    int n = n0 + row; if (n >= N) n = N - 1;
    AT[row][v] = node_attrs[(size_t)n * 4 + v];
  }

  // Async global->LDS staging of the 4 agg planes (b128 per lane, padded LDS).
  #pragma unroll
  for (int j = 0; j < 2; ++j) {
    int q   = j * 128 + tid;          // 0..255 : 16 rows x 16 quads
    int row = q >> 4, c4 = (q & 15) * 4;
    int n = n0 + row; if (n >= N) n = N - 1;
    size_t gidx = (size_t)n * 64 + c4;
    const float* g0 = aggS + gidx;
    const float* g1 = aggV + gidx;
    const float* g2 = aggV + NW + gidx;
    const float* g3 = aggV + 2 * NW + gidx;
    unsigned l0 = (unsigned)(uintptr_t)(&AS [row * PAD + c4]);
    unsigned l1 = (unsigned)(uintptr_t)(&AV0[row * PAD + c4]);
    unsigned l2 = (unsigned)(uintptr_t)(&AV1[row * PAD + c4]);
    unsigned l3 = (unsigned)(uintptr_t)(&AV2[row * PAD + c4]);
    asm volatile("global_load_async_to_lds_b128 %0, %1, off" :: "v"(l0), "v"(g0) : "memory");
    asm volatile("global_load_async_to_lds_b128 %0, %1, off" :: "v"(l1), "v"(g1) : "memory");
    asm volatile("global_load_async_to_lds_b128 %0, %1, off" :: "v"(l2), "v"(g2) : "memory");
    asm volatile("global_load_async_to_lds_b128 %0, %1, off" :: "v"(l3), "v"(g3) : "memory");
  }
  asm volatile("s_wait_asynccnt 0x0" ::: "memory");
  __syncthreads();

  const int wave  = tid >> 5;
  const int lane  = tid & 31;
  const int mrow  = lane & 15;
  const int khalf = (lane >> 4) * 2;
  const int col   = wave * 16 + (lane & 15);
  const int bbase = ((lane >> 4) << 6) + col;

  // ---- GEMM1: K=64, x = agg @ (A_LIN * Wls/Wlv) ----
  v8f xs = {0}, xv0 = {0}, xv1 = {0}, xv2 = {0};
  #pragma unroll 4
  for (int kk = 0; kk < 16; ++kk) {
    const int ab = mrow * PAD + kk * 4 + khalf;
    v2f fs = {AS[ab],  AS[ab + 1]};
    v2f f0 = {AV0[ab], AV0[ab + 1]};
    v2f f1 = {AV1[ab], AV1[ab + 1]};
    v2f f2 = {AV2[ab], AV2[ab + 1]};
    float2 wl = Wlsp[bbase + kk * 128];
    float2 wv = Wlvp[bbase + kk * 128];
    v2f bl = {wl.x, wl.y};
    v2f bv = {wv.x, wv.y};
    xs  = wmma_f32(fs, bl, xs);
    xv0 = wmma_f32(f0, bv, xv0);
    xv1 = wmma_f32(f1, bv, xv1);
    xv2 = wmma_f32(f2, bv, xv2);
  }

  // Broadcast x through LDS for the K=256 skip contraction.
  #pragma unroll
  for (int r = 0; r < 8; ++r) {
    int rowM = r + ((lane >> 4) << 3);
    XS [rowM * PAD + col] = xs[r];
    XV0[rowM * PAD + col] = xv0[r];
    XV1[rowM * PAD + col] = xv1[r];
    XV2[rowM * PAD + col] = xv2[r];
  }
  __syncthreads();

  // ---- GEMM2: K=256, A[row][u*4+v] = x[row][u] * attrs[row][v] ----
  const float atl = AT[mrow][khalf];
  const float ath = AT[mrow][khalf + 1];
  v8f ss = {0}, sv0 = {0}, sv1 = {0}, sv2 = {0};
  #pragma unroll 4
  for (int kk = 0; kk < 64; ++kk) {
    float xsv = XS [mrow * PAD + kk];
    float x0  = XV0[mrow * PAD + kk];
    float x1  = XV1[mrow * PAD + kk];
    float x2  = XV2[mrow * PAD + kk];
    v2f fa = {xsv * atl, xsv * ath};
    v2f f0 = {x0  * atl, x0  * ath};
    v2f f1 = {x1  * atl, x1  * ath};
    v2f f2 = {x2  * atl, x2  * ath};
    float2 wsp = Wssp[bbase + kk * 128];
    float2 wvp = Wsvp[bbase + kk * 128];
    v2f bs = {wsp.x, wsp.y};
    v2f bv = {wvp.x, wvp.y};
    ss  = wmma_f32(fa, bs, ss);
    sv0 = wmma_f32(f0, bv, sv0);
    sv1 = wmma_f32(f1, bv, sv1);
    sv2 = wmma_f32(f2, bv, sv2);
  }

  // ---- Epilogue: out = x + skip (A_SKIP folded), interleaved layout ----
  #pragma unroll
  for (int r = 0; r < 8; ++r) {
    int rowM = r + ((lane >> 4) << 3);
    int n = n0 + rowM;
    if (n < N) {
      float* o = out + (size_t)n * 256;
      o[col]              = xs[r]  + ss[r];
      o[64 + col * 3 + 0] = xv0[r] + sv0[r];
      o[64 + col * 3 + 1] = xv1[r] + sv1[r];
      o[64 + col * 3 + 2] = xv2[r] + sv2[r];
    }
  }
}

extern "C" void kernel_launch(void* const* d_in, const int* in_sizes, int n_in,
                              void* d_out, int out_size, void* d_ws, size_t ws_size,
                              hipStream_t stream) {
  const float*     node_feats = (const float*)d_in[0];
  const float*     node_attrs = (const float*)d_in[1];
  const float*     edge_feats = (const float*)d_in[2];
  const long long* edge_index = (const long long*)d_in[3];
  const float*     Wa  = (const float*)d_in[4];
  const float*     Wb  = (const float*)d_in[5];
  const float*     Wc  = (const float*)d_in[6];
  const float*     Wd  = (const float*)d_in[7];
  const float*     Wls = (const float*)d_in[8];
  const float*     Wlv = (const float*)d_in[9];
  const float*     Wss = (const float*)d_in[10];
  const float*     Wsv = (const float*)d_in[11];

  const int N = in_sizes[0] / 256;   // 25000
  const int E = in_sizes[2] / 4;     // 400000

  const float A_TP   = 0.0883883476483184f;   // 1/sqrt(128)
  const float IS3    = 0.5773502691896258f;   // 1/sqrt(3)
  const float A_LIN  = 0.125f;                // 1/sqrt(64)
  const float A_SKIP = 0.0625f;               // 1/sqrt(256)

  float*  aggS = (float*)d_ws;                 // [N][64]
  float*  aggV = aggS + (size_t)N * 64;        // [3][N][64]
  float2* WP   = (float2*)(aggS + (size_t)N * 256);
  float2* Wap  = WP;            // 2048 float2 each for 64x64 mats
  float2* Wbp  = WP + 2048;
  float2* Wcp  = WP + 4096;
  float2* Wdp  = WP + 6144;
  float2* Wlsp = WP + 8192;
  float2* Wlvp = WP + 10240;
  float2* Wssp = WP + 12288;    // 8192 float2 each for 256x64 mats
  float2* Wsvp = WP + 20480;

  // Workspace accumulators must start at zero on every (graph-replayed) call.
  size_t zn = (size_t)N * 256;
  zero_kernel<<<(int)((zn + 255) / 256), 256, 0, stream>>>(aggS, zn);

  // Repack weights into B-fragment order with global scales folded in.
  perm_kernel<<<16, 128, 0, stream>>>(Wa,  Wap,  32, A_TP);
  perm_kernel<<<16, 128, 0, stream>>>(Wb,  Wbp,  32, A_TP * IS3);
  perm_kernel<<<16, 128, 0, stream>>>(Wc,  Wcp,  32, A_TP);
  perm_kernel<<<16, 128, 0, stream>>>(Wd,  Wdp,  32, A_TP);
  perm_kernel<<<16, 128, 0, stream>>>(Wls, Wlsp, 32, A_LIN);
  perm_kernel<<<16, 128, 0, stream>>>(Wlv, Wlvp, 32, A_LIN);
  perm_kernel<<<64, 128, 0, stream>>>(Wss, Wssp, 128, A_SKIP);
  perm_kernel<<<64, 128, 0, stream>>>(Wsv, Wsvp, 128, A_SKIP);

  int etiles = (E + 15) / 16;
  edge_kernel<<<etiles, 128, 0, stream>>>(node_feats, edge_feats, edge_index,
                                          Wap, Wbp, Wcp, Wdp, aggS, aggV, N, E);

  int ntiles = (N + 15) / 16;
  node_kernel<<<ntiles, 128, 0, stream>>>(node_attrs, Wlsp, Wlvp, Wssp, Wsvp,
                                          aggS, aggV, (float*)d_out, N);
}